// ListwiseSoftmaxLoss_88476326298378
// MI455X (gfx1250) — compile-verified
//
#include <hip/hip_runtime.h>
#include <float.h>

#define WAVE 32
#define MAXSEG 512
#define LDSPAD 4
#define WAVES_PER_BLOCK 8
#define BLOCK_THREADS (WAVES_PER_BLOCK * WAVE)
#define NBLOCKS 4096
#define WS_STARTS_OFF 16   // bytes reserved for [total, nvalid] accumulators

typedef int v4i_t __attribute__((vector_size(4 * sizeof(int))));

// ---- CDNA5 async global->LDS copies (ASYNCcnt path), with compiling fallbacks ----
#if defined(__gfx1250__) && __has_builtin(__builtin_amdgcn_global_load_async_to_lds_b32)
#define ASYNC_COPY_B32(ldst, gsrc)                                             \
  __builtin_amdgcn_global_load_async_to_lds_b32(                               \
      (__attribute__((address_space(1))) int*)(void*)(gsrc),                   \
      (__attribute__((address_space(3))) int*)(void*)(ldst), 0, 0)
#if __has_builtin(__builtin_amdgcn_s_wait_asynccnt)
#define ASYNC_WAIT() __builtin_amdgcn_s_wait_asynccnt(0)
#else
#define ASYNC_WAIT() asm volatile("s_wait_asynccnt 0" ::: "memory")
#endif
#else
#define ASYNC_COPY_B32(ldst, gsrc) ((void)(*(ldst) = *(gsrc)))
#define ASYNC_WAIT() ((void)0)
#endif

#if defined(__gfx1250__) && __has_builtin(__builtin_amdgcn_global_load_async_to_lds_b128)
#define ASYNC_COPY_B128(ldst, gsrc)                                            \
  __builtin_amdgcn_global_load_async_to_lds_b128(                              \
      (__attribute__((address_space(1))) v4i_t*)(void*)(gsrc),                 \
      (__attribute__((address_space(3))) v4i_t*)(void*)(ldst), 0, 0)
#else
#define ASYNC_COPY_B128(ldst, gsrc)                                            \
  do {                                                                         \
    ASYNC_COPY_B32((ldst) + 0, (gsrc) + 0);                                    \
    ASYNC_COPY_B32((ldst) + 1, (gsrc) + 1);                                    \
    ASYNC_COPY_B32((ldst) + 2, (gsrc) + 2);                                    \
    ASYNC_COPY_B32((ldst) + 3, (gsrc) + 3);                                    \
  } while (0)
#endif

__device__ __forceinline__ int lower_bound_i32(const int* __restrict__ a, int n, int v) {
  int lo = 0, hi = n;
  while (lo < hi) {
    int mid = (int)(((unsigned)lo + (unsigned)hi) >> 1);
    if (a[mid] < v) lo = mid + 1; else hi = mid;
  }
  return lo;
}

__device__ __forceinline__ float wave_max_f32(float v) {
#pragma unroll
  for (int off = 16; off > 0; off >>= 1) v = fmaxf(v, __shfl_xor(v, off, 32));
  return v;
}

__device__ __forceinline__ float wave_sum_f32(float v) {
#pragma unroll
  for (int off = 16; off > 0; off >>= 1) v += __shfl_xor(v, off, 32);
  return v;
}

// Three-pass per-segment KL. Inlined separately at each call site so the
// address space (LDS vs global) of sp/lp is inferred -> ds_load vs global_load.
__device__ __forceinline__ float segment_kl(const float* sp, const float* lp,
                                            int count, int lane) {
  float ml = -FLT_MAX, ms = -FLT_MAX;
  for (int k = lane; k < count; k += WAVE) {
    ml = fmaxf(ml, lp[k]);
    ms = fmaxf(ms, sp[k]);
  }
  ml = wave_max_f32(ml);
  ms = wave_max_f32(ms);
  float dl = 0.0f, dsum = 0.0f;
  for (int k = lane; k < count; k += WAVE) {
    dl   += __expf(lp[k] - ml);
    dsum += __expf(sp[k] - ms);
  }
  dl   = wave_sum_f32(dl);
  dsum = wave_sum_f32(dsum);
  const float invdl  = 1.0f / dl;
  const float logds  = __logf(dsum);
  const float smooth = 0.1f / (float)count;
  float kl = 0.0f;
  for (int k = lane; k < count; k += WAVE) {
    const float t    = 0.9f * __expf(lp[k] - ml) * invdl + smooth;
    const float logp = (sp[k] - ms) - logds;
    kl += t * (__logf(t) - logp);
  }
  return wave_sum_f32(kl);
}

// Streaming boundary pass: starts[q] = lower_bound(qids, q) for q in [0, nq].
// Coalesced single read of qids (~3us at 23.3 TB/s) replaces 131K serialized
// binary searches (latency-bound pointer chase) in the main kernel.
__global__ void __launch_bounds__(256)
boundary_kernel(const int* __restrict__ qids, const int* __restrict__ d_nq,
                int n, int* __restrict__ starts, long long cap_bytes) {
  const int nq = d_nq[0];
  if ((long long)(nq + 1) * 4 > cap_bytes) return;   // scratch too small: skip
  const int i = blockIdx.x * blockDim.x + threadIdx.x;
  if (i >= n) return;
  const int cur  = qids[i];
  const int prev = (i == 0) ? -1 : qids[i - 1];
  for (int q = prev + 1; q <= cur; ++q) starts[q] = i;   // fills gaps (empty queries)
  if (i == n - 1) {
    for (int q = cur + 1; q <= nq; ++q) starts[q] = n;
  }
}

__global__ void __launch_bounds__(BLOCK_THREADS)
listwise_kl_kernel(const float* __restrict__ scores,
                   const float* __restrict__ labels,
                   const int*   __restrict__ qids,
                   const int*   __restrict__ d_nq,
                   int n,
                   float* __restrict__ ws,
                   const int* __restrict__ starts,
                   long long cap_bytes) {
  __shared__ float s_sco[WAVES_PER_BLOCK][MAXSEG + LDSPAD];
  __shared__ float s_lab[WAVES_PER_BLOCK][MAXSEG + LDSPAD];

  const int nq     = d_nq[0];
  const bool preOk = ((long long)(nq + 1) * 4 <= cap_bytes);
  const int w      = threadIdx.x >> 5;
  const int lane   = threadIdx.x & 31;
  const int gwave  = blockIdx.x * WAVES_PER_BLOCK + w;
  const int nwav   = gridDim.x * WAVES_PER_BLOCK;

  float acc_total  = 0.0f;
  float acc_nvalid = 0.0f;

  float* const sd = s_sco[w];
  float* const ld = s_lab[w];

  for (int q = gwave; q < nq; q += nwav) {
    int start, end;
    if (preOk) {
      start = starts[q];
      end   = starts[q + 1];
    } else {
      start = lower_bound_i32(qids, n, q);
      end   = lower_bound_i32(qids, n, q + 1);
    }
    const int count = end - start;
    if (count < 2) continue;   // reference skips queries with < 2 docs

    const int  shift = start & 3;            // LDS skew so b128 is 16B/16B aligned
    const bool doLds = (count <= MAXSEG);

    // ---- Kick async DMA for this query's segment (global -> LDS) ----
    if (doLds) {
      const float* gsco = scores + start;
      const float* glab = labels + start;
      const int head = (4 - shift) & 3;      // elems before 16B-aligned boundary
      const int hc   = head < count ? head : count;
      for (int k = lane; k < hc; k += WAVE) {   // <=3 elems, lanes 0..2
        ASYNC_COPY_B32(sd + shift + k, gsco + k);
        ASYNC_COPY_B32(ld + shift + k, glab + k);
      }
      if (count > head) {
        const int nvec = (count - head) >> 2;   // full 4-elem groups
        for (int t = lane; t < nvec; t += WAVE) {
          const int k = head + 4 * t;
          ASYNC_COPY_B128(sd + shift + k, gsco + k);
          ASYNC_COPY_B128(ld + shift + k, glab + k);
        }
        for (int k = head + 4 * nvec + lane; k < count; k += WAVE) {
          ASYNC_COPY_B32(sd + shift + k, gsco + k);
          ASYNC_COPY_B32(ld + shift + k, glab + k);
        }
      }
    }

    // ---- Overlap: warm L2 with the NEXT query's segment while DMA flies ----
    const int qn = q + nwav;
    if (preOk && qn < nq) {
      const int sn = starts[qn];
      __builtin_prefetch(scores + sn + lane * 4, 0, 0);   // global_prefetch_b8
      __builtin_prefetch(labels + sn + lane * 4, 0, 0);   // 512B span per array
    }

    // ---- Consume this query's segment ----
    float kl;
    if (doLds) {
      ASYNC_WAIT();
      kl = segment_kl(sd + shift, ld + shift, count, lane);   // ds_load path
    } else {
      // Astronomically rare (Binomial(16.7M, 1/131072) tail); L2 re-reads.
      kl = segment_kl(scores + start, labels + start, count, lane);
    }
    acc_total  += kl;
    acc_nvalid += 1.0f;
  }

  if (lane == 0 && acc_nvalid > 0.0f) {
    atomicAdd(&ws[0], acc_total);
    atomicAdd(&ws[1], acc_nvalid);
  }
}

__global__ void init_ws_kernel(float* __restrict__ ws) {
  if (threadIdx.x == 0 && blockIdx.x == 0) {
    ws[0] = 0.0f;
    ws[1] = 0.0f;
  }
}

__global__ void finalize_kernel(const float* __restrict__ ws, float* __restrict__ out) {
  if (threadIdx.x == 0 && blockIdx.x == 0) {
    out[0] = ws[0] / fmaxf(ws[1], 1.0f);
  }
}

extern "C" void kernel_launch(void* const* d_in, const int* in_sizes, int n_in,
                              void* d_out, int out_size, void* d_ws, size_t ws_size,
                              hipStream_t stream) {
  const float* scores = (const float*)d_in[0];
  const float* labels = (const float*)d_in[1];
  const int*   qids   = (const int*)d_in[2];
  const int*   d_nq   = (const int*)d_in[3];   // num_queries (device scalar)
  const int    n      = in_sizes[0];
  float*       ws     = (float*)d_ws;
  float*       out    = (float*)d_out;

  int*      starts    = (int*)((char*)d_ws + WS_STARTS_OFF);
  long long cap_bytes = (ws_size > (size_t)WS_STARTS_OFF)
                          ? (long long)(ws_size - WS_STARTS_OFF) : 0;

  init_ws_kernel<<<1, 32, 0, stream>>>(ws);
  boundary_kernel<<<(n + 255) / 256, 256, 0, stream>>>(qids, d_nq, n, starts, cap_bytes);
  listwise_kl_kernel<<<NBLOCKS, BLOCK_THREADS, 0, stream>>>(scores, labels, qids, d_nq, n,
                                                            ws, starts, cap_bytes);
  finalize_kernel<<<1, 32, 0, stream>>>(ws, out);
}